// custom2D_20598663151923
// MI455X (gfx1250) — compile-verified
//
#include <hip/hip_runtime.h>

// OCR letter conv as WMMA GEMM for gfx1250 (MI455X).
// out[letter, o] = sum_k M[o,k] * x[letter,k],  M built from 3x3 kernel.
// fp32 WMMA 16x16x4 keeps reference precision; workload is HBM-bound
// (~222 MB @ 23.3 TB/s ~= 9.6 us). X tiles are staged into LDS with the
// CDNA5 async global->LDS path (no VGPR round trip), fragments are read
// with conflict-free ds_load_2addr_b64 pairs, and output uses NT stores.

typedef float v2f __attribute__((ext_vector_type(2)));
typedef float v8f __attribute__((ext_vector_type(8)));

#define LH 16
#define LW 8
#define KDIM 128          // 16*8 pixels per letter
#define OW 6
#define ODIM 84           // 14*6 conv outputs per letter
#define OPAD 96           // padded to 6 tiles of 16 rows
#define MSTR 132          // LDS row stride (floats): 132 mod 64 banks = 4 -> conflict-free
#define WAVES 8
#define LPW 16            // letters per wave
#define LPB (WAVES * LPW) // letters per block = 128

__global__ __launch_bounds__(256) void conv_letters_wmma(
    const float* __restrict__ x, const float* __restrict__ kern,
    float* __restrict__ out)
{
    __shared__ float Ms[OPAD * MSTR];        // conv-as-matmul matrix M[o][k]
    __shared__ float Xs[WAVES * LPW * MSTR]; // per-wave letter staging

    const int tid  = threadIdx.x;
    const int lane = tid & 31;
    const int wave = tid >> 5;
    const int ml   = lane & 15;  // row/col within 16
    const int hi   = lane >> 4;  // lane half selects K pair

    const int lb = blockIdx.x * LPB + wave * LPW; // first letter of this wave
    float* xw = &Xs[wave * (LPW * MSTR)];

    // ---- Kick off async global->LDS staging of this wave's 16 letters ----
    // Each instruction: 32 lanes x 16B = 512B, one full letter, coalesced.
    // ASYNCcnt-tracked; no VGPR data round trip.
    #pragma unroll
    for (int i = 0; i < LPW; ++i) {
        const unsigned long long gaddr =
            (unsigned long long)(size_t)(x + (size_t)(lb + i) * KDIM + 4 * lane);
        const unsigned ldsaddr =
            (unsigned)(size_t)(xw + i * MSTR + 4 * lane); // low 32 bits = LDS byte addr
        asm volatile("global_load_async_to_lds_b128 %0, %1, off"
                     :: "v"(ldsaddr), "v"(gaddr)
                     : "memory");
    }

    // ---- Build M[o][k] in LDS while the async DMA streams X ----
    for (int idx = tid; idx < OPAD * KDIM; idx += 256) {
        const int o = idx / KDIM;
        const int k = idx - o * KDIM;
        float v = 0.0f;
        if (o < ODIM) {
            const int oh = o / OW, ow = o - oh * OW;
            const int ih = k >> 3, iw = k & 7;
            const int dh = ih - oh, dw = iw - ow;
            if (dh >= 0 && dh < 3 && dw >= 0 && dw < 3)
                v = kern[dh * 3 + dw];
        }
        Ms[o * MSTR + k] = v;
    }
    __syncthreads();

    // Wait for this wave's async X staging before touching fragments.
    asm volatile("s_wait_asynccnt 0x0" ::: "memory");

    // ---- GEMM: D(16 letters x 96 outs) = A(16x128) * B(128x96) ----
    v8f acc[6];
    #pragma unroll
    for (int r = 0; r < 6; ++r) acc[r] = v8f{};

    // A fragment (16x4 f32): lane l holds row ml, K = 4s + 2*hi (+1)
    const float* aBase = xw + ml * MSTR + 2 * hi;
    // B fragment (4x16 f32): lane l holds col ml (out row r*16+ml), same K pair
    const float* bBase = &Ms[ml * MSTR + 2 * hi];

    for (int s = 0; s < KDIM / 4; ++s) {
        const v2f a = *(const v2f*)(aBase + 4 * s);
        #pragma unroll
        for (int r = 0; r < 6; ++r) {
            const v2f b = *(const v2f*)(bBase + (size_t)(r * 16) * MSTR + 4 * s);
            acc[r] = __builtin_amdgcn_wmma_f32_16x16x4_f32(
                /*neg_a=*/false, a, /*neg_b=*/false, b,
                /*c_mod=*/(short)0, acc[r],
                /*reuse_a=*/false, /*reuse_b=*/false);
        }
    }

    // ---- Store D: VGPR i -> letter lb+i+8*hi, column o = r*16 + ml ----
    // lanes 0..15 write consecutive o -> 64B contiguous per half-wave.
    // Output is written once and never re-read: non-temporal stores.
    #pragma unroll
    for (int r = 0; r < 6; ++r) {
        const int o = r * 16 + ml;
        if (o < ODIM) {
            #pragma unroll
            for (int i = 0; i < 8; ++i) {
                __builtin_nontemporal_store(
                    acc[r][i], &out[(size_t)(lb + i + 8 * hi) * ODIM + o]);
            }
        }
    }
}

extern "C" void kernel_launch(void* const* d_in, const int* in_sizes, int n_in,
                              void* d_out, int out_size, void* d_ws, size_t ws_size,
                              hipStream_t stream) {
    const float* x    = (const float*)d_in[0]; // (512, 512, 128) f32
    const float* kern = (const float*)d_in[1]; // (3, 3) f32
    float* out = (float*)d_out;                // (1, 512, 512, 84) f32

    const int nLetters = in_sizes[0] / KDIM;   // 262144
    const int blocks   = nLetters / LPB;       // 2048 (divides exactly)

    hipLaunchKernelGGL(conv_letters_wmma, dim3(blocks), dim3(256), 0, stream,
                       x, kern, out);
}